// MultiLoRA_48524540510408
// MI455X (gfx1250) — compile-verified
//
#include <hip/hip_runtime.h>

// MultiLoRA fused kernel for MI455X (gfx1250, wave32).
// out = x @ W + (x @ lora_A[idx]) @ lora_B[idx] * SCALING
// HBM-bound (576 MB min traffic vs 0.55 TFLOP @ 23.3 TB/s) -> stay f32,
// V_WMMA_F32_16X16X4_F32 for all matrix math, single pass over x (LoRA
// rank-16 projection fused into the K loop), and CDNA5 async-to-LDS
// copies (ASYNCcnt) with double-buffered LDS for the tile pipeline.

typedef float v2f __attribute__((ext_vector_type(2)));
typedef float v8f __attribute__((ext_vector_type(8)));

#define D_IN    4096
#define D_OUT   4096
#define RANK    16
#define S_LEN   2048
#define M_TILE  128
#define N_TILE  128
#define K_TILE  32
#define XPITCH  36     // x tile pitch: conflict-free 16-lane strides + 16B align
#define WPITCH  136    // W tile pitch: disjoint bank windows for half-waves
#define EPITCH  20     // xr tile pitch
#define SCALING 1.0f   // 16.0 / 16

// CDNA5 async tensor path: direct global -> LDS, tracked by ASYNCcnt.
#define ASYNC_B128(ldsoff, gptr)                                              \
    asm volatile("global_load_async_to_lds_b128 %0, %1, off"                  \
                 :: "v"((unsigned)(ldsoff)),                                  \
                    "v"((unsigned long long)(gptr)) : "memory")
#define ASYNC_B64(ldsoff, gptr)                                               \
    asm volatile("global_load_async_to_lds_b64 %0, %1, off"                   \
                 :: "v"((unsigned)(ldsoff)),                                  \
                    "v"((unsigned long long)(gptr)) : "memory")

__launch_bounds__(256)
__global__ void multilora_wmma_kernel(const float* __restrict__ x,
                                      const int*   __restrict__ adapter_indices,
                                      const float* __restrict__ W,
                                      const float* __restrict__ lora_A,
                                      const float* __restrict__ lora_B,
                                      float* __restrict__ out)
{
    __shared__ __align__(16) float sx2[2][M_TILE * XPITCH];  // x tile [m][k]
    __shared__ __align__(16) float sw2[2][K_TILE * WPITCH];  // W tile [k][n]
    __shared__ __align__(16) float sA2[2][K_TILE * RANK];    // lora_A tile [k][r]
    __shared__ __align__(16) float sxr[M_TILE * EPITCH];     // xr = x @ A[a]
    __shared__ __align__(16) float slB[RANK * WPITCH];       // lora_B tile [r][n]

    const int tid  = threadIdx.x;
    const int lane = tid & 31;
    const int wid  = tid >> 5;       // 8 waves
    const int wm   = wid & 3;        // 4 row-strips of 32
    const int wn   = wid >> 2;       // 2 col-strips of 64
    const int l    = lane & 15;      // lane within half-wave
    const int hi   = lane >> 4;      // 0 -> (k,k+1), 1 -> (k+2,k+3)

    const int n0 = blockIdx.x * N_TILE;
    const int m0 = blockIdx.y * M_TILE;
    const int b  = m0 >> 11;                     // M_TILE | S_LEN -> one batch/tile
    const int a  = adapter_indices[b];           // uniform scalar load

    const float* lA = lora_A + (size_t)a * D_IN * RANK;
    const float* lB = lora_B + (size_t)a * RANK * D_OUT;

    const int rm = wm * 32;
    const int cn = wn * 64;

    v8f acc[2][4];
    v8f xr[2];
    {
        const v8f z = {0.f,0.f,0.f,0.f,0.f,0.f,0.f,0.f};
        for (int mf = 0; mf < 2; ++mf) {
            xr[mf] = z;
            for (int nf = 0; nf < 4; ++nf) acc[mf][nf] = z;
        }
    }

    // ---- async staging geometry (256 threads, b128 lanes) ----
    const int xrow = tid >> 3;           // 0..31 (+32*rr)
    const int xcol = (tid & 7) * 4;      // 0..28
    const int wk   = tid >> 3;           // 0..31 (k row of W)
    const int wc   = (tid & 7) * 4;      // 0..28 (+32*rr)

    // 9 async issues per wave per tile: 4x b128 (x) + 4x b128 (W) + 1x b64 (lora_A)
    auto issue_tile = [&](int k0, int bufsel) {
        #pragma unroll
        for (int rr = 0; rr < 4; ++rr) {
            unsigned lo = (unsigned)(size_t)&sx2[bufsel][(xrow + 32*rr) * XPITCH + xcol];
            const float* gp = &x[(size_t)(m0 + xrow + 32*rr) * D_IN + k0 + xcol];
            ASYNC_B128(lo, gp);
        }
        #pragma unroll
        for (int rr = 0; rr < 4; ++rr) {
            unsigned lo = (unsigned)(size_t)&sw2[bufsel][wk * WPITCH + wc + 32*rr];
            const float* gp = &W[(size_t)(k0 + wk) * D_OUT + n0 + wc + 32*rr];
            ASYNC_B128(lo, gp);
        }
        {
            unsigned lo = (unsigned)(size_t)&sA2[bufsel][2 * tid];
            const float* gp = &lA[(size_t)k0 * RANK + 2 * tid];
            ASYNC_B64(lo, gp);
        }
    };

    issue_tile(0, 0);

    const int KITERS = D_IN / K_TILE;   // 128
    for (int kt = 0; kt < KITERS; ++kt) {
        const int buf = kt & 1;
        __syncthreads();   // other buffer's readers retired before overwrite
        if (kt + 1 < KITERS) {
            issue_tile(kt + 1, buf ^ 1);
            asm volatile("s_wait_asynccnt 9" ::: "memory");  // current tile landed
        } else {
            asm volatile("s_wait_asynccnt 0" ::: "memory");
        }
        __syncthreads();   // publish current buffer to all waves

        // 8 k-steps of V_WMMA_F32_16X16X4_F32
        #pragma unroll
        for (int ks = 0; ks < K_TILE; ks += 4) {
            v2f af[2], bf[4], lf;
            #pragma unroll
            for (int mf = 0; mf < 2; ++mf) {   // contiguous pair -> ds_load_b64
                const float* p = &sx2[buf][(rm + mf*16 + l) * XPITCH + ks + 2*hi];
                v2f t = {p[0], p[1]};
                af[mf] = t;
            }
            #pragma unroll
            for (int nf = 0; nf < 4; ++nf) {   // strided pair -> ds_load_2addr_b32
                const float* p = &sw2[buf][(ks + 2*hi) * WPITCH + cn + nf*16 + l];
                v2f t = {p[0], p[WPITCH]};
                bf[nf] = t;
            }
            {
                const float* p = &sA2[buf][(ks + 2*hi) * RANK + l];
                v2f t = {p[0], p[RANK]};
                lf = t;
            }
            #pragma unroll
            for (int mf = 0; mf < 2; ++mf)
                #pragma unroll
                for (int nf = 0; nf < 4; ++nf)
                    acc[mf][nf] = __builtin_amdgcn_wmma_f32_16x16x4_f32(
                        false, af[mf], false, bf[nf], (short)0, acc[mf][nf], false, false);
            #pragma unroll
            for (int mf = 0; mf < 2; ++mf)
                xr[mf] = __builtin_amdgcn_wmma_f32_16x16x4_f32(
                    false, af[mf], false, lf, (short)0, xr[mf], false, false);
        }
    }

    // ---- LoRA epilogue: acc += xr @ lora_B[a] (K = RANK = 16) ----
    // spill xr (D-layout) to LDS; wn==0/1 waves computed identical xr.
    if (wn == 0) {
        #pragma unroll
        for (int mf = 0; mf < 2; ++mf) {
            const int rbase = rm + mf*16 + 8*hi;
            #pragma unroll
            for (int v = 0; v < 8; ++v)
                sxr[(rbase + v) * EPITCH + l] = xr[mf][v] * SCALING;
        }
    }
    // stage lora_B tile k-major: slB[r][n]
    {
        const int k = tid >> 4;            // 0..15
        const int c = (tid & 15) * 4;      // 0..60
        for (int rr = 0; rr < 2; ++rr) {
            float4 v = *(const float4*)&lB[(size_t)k * D_OUT + n0 + c + 64*rr];
            *(float4*)&slB[k * WPITCH + c + 64*rr] = v;
        }
    }
    __syncthreads();

    #pragma unroll
    for (int ks = 0; ks < RANK; ks += 4) {
        v2f af[2], bf[4];
        #pragma unroll
        for (int mf = 0; mf < 2; ++mf) {
            const float* p = &sxr[(rm + mf*16 + l) * EPITCH + ks + 2*hi];
            v2f t = {p[0], p[1]};
            af[mf] = t;
        }
        #pragma unroll
        for (int nf = 0; nf < 4; ++nf) {
            const float* p = &slB[(ks + 2*hi) * WPITCH + cn + nf*16 + l];
            v2f t = {p[0], p[WPITCH]};
            bf[nf] = t;
        }
        #pragma unroll
        for (int mf = 0; mf < 2; ++mf)
            #pragma unroll
            for (int nf = 0; nf < 4; ++nf)
                acc[mf][nf] = __builtin_amdgcn_wmma_f32_16x16x4_f32(
                    false, af[mf], false, bf[nf], (short)0, acc[mf][nf], false, false);
    }

    // ---- store (D-layout: lanes 0-15 -> M=v, lanes 16-31 -> M=v+8; N=l) ----
    #pragma unroll
    for (int mf = 0; mf < 2; ++mf) {
        #pragma unroll
        for (int nf = 0; nf < 4; ++nf) {
            const int row0 = m0 + rm + mf*16 + 8*hi;
            const int col  = n0 + cn + nf*16 + l;
            float* o = &out[(size_t)row0 * D_OUT + col];
            #pragma unroll
            for (int v = 0; v < 8; ++v)
                o[(size_t)v * D_OUT] = acc[mf][nf][v];
        }
    }
}

extern "C" void kernel_launch(void* const* d_in, const int* in_sizes, int n_in,
                              void* d_out, int out_size, void* d_ws, size_t ws_size,
                              hipStream_t stream) {
    const float* x   = (const float*)d_in[0];
    const int*   idx = (const int*)  d_in[1];
    const float* W   = (const float*)d_in[2];
    const float* lA  = (const float*)d_in[3];
    const float* lB  = (const float*)d_in[4];
    float* out = (float*)d_out;

    dim3 grid(D_OUT / N_TILE, (8 * S_LEN) / M_TILE);  // (32, 128)
    multilora_wmma_kernel<<<grid, 256, 0, stream>>>(x, idx, W, lA, lB, out);
}